// GroupNorm_42631845380831
// MI455X (gfx1250) — compile-verified
//
#include <hip/hip_runtime.h>
#include <hip/hip_bf16.h>

// Problem constants (match the reference)
#define GN_B   128
#define GN_G   32
#define GN_GS  8
#define GN_C   256
#define GN_EPS 1e-5f

// Workspace layout (float offsets):
//   [0      , 4096 )  sum   [B*G]
//   [4096   , 8192 )  sumsq [B*G]
//   [8192   , 8320 )  cnt   [B]
//   [8320   , 12416)  scale [B*G]
//   [12416  , 16512)  shift [B*G]
#define WS_SUM   0
#define WS_SQ    (GN_B * GN_G)
#define WS_CNT   (2 * GN_B * GN_G)
#define WS_SCALE (2 * GN_B * GN_G + GN_B)
#define WS_SHIFT (2 * GN_B * GN_G + GN_B + GN_B * GN_G)
#define WS_ZERO_COUNT (2 * GN_B * GN_G + GN_B)   // sum+sq+cnt must be zeroed

// -------- Pass 0: zero the accumulators (harness poisons ws with 0xAA) -----
__global__ void gn_zero(float* __restrict__ ws) {
  for (int i = blockIdx.x * blockDim.x + threadIdx.x; i < WS_ZERO_COUNT;
       i += gridDim.x * blockDim.x)
    ws[i] = 0.0f;
}

// -------- Pass 1: per-(graph,group) sum / sumsq / count ---------------------
// Wave-centric: lane = group id; each wave owns a contiguous node range.
// batch is sorted, so flush atomics only on graph-id change.
__global__ void gn_stats(const float4* __restrict__ x4,
                         const long long* __restrict__ batch,
                         float* __restrict__ ws, int N, int nodes_per_wave) {
  float* sum = ws + WS_SUM;
  float* sq  = ws + WS_SQ;
  float* cnt = ws + WS_CNT;

  const int wave = blockIdx.x * (blockDim.x >> 5) + (threadIdx.x >> 5);
  const int lane = threadIdx.x & 31;                 // lane == group id
  long long start = (long long)wave * nodes_per_wave;
  long long end   = start + nodes_per_wave;
  if (end > N) end = N;
  if (start >= end) return;

  float s = 0.0f, ss = 0.0f, run = 0.0f;
  int cur = (int)batch[start];

  for (long long n = start; n < end; ++n) {
    const int b = (int)batch[n];
    if (b != cur) {                                  // uniform across the wave
      atomicAdd(&sum[cur * GN_G + lane], s);
      atomicAdd(&sq [cur * GN_G + lane], ss);
      if (lane == 0) atomicAdd(&cnt[cur], run);
      s = 0.0f; ss = 0.0f; run = 0.0f; cur = b;
    }
    // lane covers channels [8*lane, 8*lane+8): two coalesced b128 loads
    const float4 a = x4[n * (GN_C / 4) + lane * 2 + 0];
    const float4 c = x4[n * (GN_C / 4) + lane * 2 + 1];
    s  += ((a.x + a.y) + (a.z + a.w)) + ((c.x + c.y) + (c.z + c.w));
    ss = fmaf(a.x, a.x, ss); ss = fmaf(a.y, a.y, ss);
    ss = fmaf(a.z, a.z, ss); ss = fmaf(a.w, a.w, ss);
    ss = fmaf(c.x, c.x, ss); ss = fmaf(c.y, c.y, ss);
    ss = fmaf(c.z, c.z, ss); ss = fmaf(c.w, c.w, ss);
    run += 1.0f;
  }
  atomicAdd(&sum[cur * GN_G + lane], s);
  atomicAdd(&sq [cur * GN_G + lane], ss);
  if (lane == 0) atomicAdd(&cnt[cur], run);
}

// -------- Pass 2: fold stats + weight/bias into per-(b,g) scale/shift -------
__global__ void gn_finalize(float* __restrict__ ws,
                            const float* __restrict__ wgt,
                            const float* __restrict__ bias) {
  const int i = blockIdx.x * blockDim.x + threadIdx.x;
  if (i >= GN_B * GN_G) return;
  const int b = i / GN_G;
  const float norm = fmaxf(ws[WS_CNT + b], 1.0f) * (float)GN_GS;
  const float mean = ws[WS_SUM + i] / norm;
  float var = ws[WS_SQ + i] / norm - mean * mean;
  var = fmaxf(var, 0.0f);
  const float inv = rsqrtf(var + GN_EPS);
  const float sc  = inv * wgt[0];
  ws[WS_SCALE + i] = sc;
  ws[WS_SHIFT + i] = fmaf(-mean, sc, bias[0]);
}

// -------- Pass 3: out = x*scale[b,g] + shift[b,g] ---------------------------
// CDNA5 async path: each wave double-buffers its own 1KB rows through LDS via
// global_load_async_to_lds_b128 (ASYNCcnt), then ds_load_b128 -> FMA -> store.
#if defined(__has_builtin)
#if __has_builtin(__builtin_amdgcn_global_load_async_to_lds_b128)
#define GN_HAS_ASYNC 1
#endif
#endif
#ifndef GN_HAS_ASYNC
#define GN_HAS_ASYNC 0
#endif

// The builtin wants typed int4 pointers (per hipcc diagnostic):
//   arg0: v4i in global address space, arg1: v4i in LDS address space
typedef int v4i __attribute__((ext_vector_type(4)));
typedef __attribute__((address_space(1))) v4i gv4i_t;   // global (64-bit ptr)
typedef __attribute__((address_space(3))) v4i lv4i_t;   // LDS    (32-bit ptr)

__global__ void gn_apply(const float* __restrict__ x,
                         const long long* __restrict__ batch,
                         const float* __restrict__ ws,
                         float* __restrict__ out, int N, int rows_per_wave) {
  const float* scale = ws + WS_SCALE;
  const float* shift = ws + WS_SHIFT;
  const int w    = threadIdx.x >> 5;
  const int lane = threadIdx.x & 31;

  long long base = ((long long)blockIdx.x * 8 + w) * rows_per_wave;
  if (base >= N) return;
  int nrows = rows_per_wave;
  if (base + nrows > N) nrows = (int)(N - base);

#if GN_HAS_ASYNC
  // [wave][buffer][row of 256 floats] -> 16KB per 256-thread block.
  // Wave w's 32 lanes (32B each) exactly cover wave w's own row: no barrier.
  __shared__ float lds[8][2][GN_C];

  auto issue = [&](long long r, int buf) {
    const float* gsrc = x + r * GN_C + lane * 8;     // 32B per lane
    gv4i_t* g = (gv4i_t*)(unsigned long long)gsrc;
    lv4i_t* l = (lv4i_t*)(unsigned int)(unsigned long long)(&lds[w][buf][lane * 8]);
    // INST_OFFSET is added to BOTH global and LDS addresses (ISA 08 §4.4)
    __builtin_amdgcn_global_load_async_to_lds_b128(g, l, 0, 0);
    __builtin_amdgcn_global_load_async_to_lds_b128(g, l, 16, 0);
  };

  issue(base, 0);
  if (nrows > 1) issue(base + 1, 1);

  for (int i = 0; i < nrows; ++i) {
    // Async loads complete in order: <=2 outstanding means row i has landed.
    if (i + 1 < nrows) asm volatile("s_wait_asynccnt 0x2" ::: "memory");
    else               asm volatile("s_wait_asynccnt 0x0" ::: "memory");

    const long long r = base + i;
    const int b = (int)batch[r];
    const float sc = scale[b * GN_G + lane];
    const float sh = shift[b * GN_G + lane];

    const float4* lb = (const float4*)&lds[w][i & 1][0];
    const float4 a = lb[lane * 2 + 0];
    const float4 c = lb[lane * 2 + 1];
    float4 oa, oc;
    oa.x = fmaf(a.x, sc, sh); oa.y = fmaf(a.y, sc, sh);
    oa.z = fmaf(a.z, sc, sh); oa.w = fmaf(a.w, sc, sh);
    oc.x = fmaf(c.x, sc, sh); oc.y = fmaf(c.y, sc, sh);
    oc.z = fmaf(c.z, sc, sh); oc.w = fmaf(c.w, sc, sh);

    float4* o4 = (float4*)(out + r * GN_C);
    o4[lane * 2 + 0] = oa;
    o4[lane * 2 + 1] = oc;

    if (i + 2 < nrows) {
      // Make sure this buffer's ds reads retired before the DMA rewrites it.
      asm volatile("s_wait_dscnt 0x0" ::: "memory");
      issue(base + i + 2, i & 1);
    }
  }
#else
  // Fallback: plain coalesced b128 streaming (still bandwidth-optimal).
  const float4* x4 = (const float4*)x;
  for (int i = 0; i < nrows; ++i) {
    const long long r = base + i;
    const int b = (int)batch[r];
    const float sc = scale[b * GN_G + lane];
    const float sh = shift[b * GN_G + lane];
    const float4 a = x4[r * (GN_C / 4) + lane * 2 + 0];
    const float4 c = x4[r * (GN_C / 4) + lane * 2 + 1];
    float4 oa, oc;
    oa.x = fmaf(a.x, sc, sh); oa.y = fmaf(a.y, sc, sh);
    oa.z = fmaf(a.z, sc, sh); oa.w = fmaf(a.w, sc, sh);
    oc.x = fmaf(c.x, sc, sh); oc.y = fmaf(c.y, sc, sh);
    oc.z = fmaf(c.z, sc, sh); oc.w = fmaf(c.w, sc, sh);
    float4* o4 = (float4*)(out + r * GN_C);
    o4[lane * 2 + 0] = oa;
    o4[lane * 2 + 1] = oc;
  }
#endif
}

// ---------------------------------------------------------------------------
extern "C" void kernel_launch(void* const* d_in, const int* in_sizes, int n_in,
                              void* d_out, int out_size, void* d_ws, size_t ws_size,
                              hipStream_t stream) {
  const float*     x     = (const float*)d_in[0];
  const long long* batch = (const long long*)d_in[1];   // jnp.int64
  const float*     wgt   = (const float*)d_in[2];
  const float*     bias  = (const float*)d_in[3];
  float* ws  = (float*)d_ws;
  float* out = (float*)d_out;
  const int N = in_sizes[1];

  gn_zero<<<33, 256, 0, stream>>>(ws);

  const int blocks1 = 1024;                    // 8192 waves
  const int waves1  = blocks1 * 8;
  const int npw     = (N + waves1 - 1) / waves1;
  gn_stats<<<blocks1, 256, 0, stream>>>((const float4*)x, batch, ws, N, npw);

  gn_finalize<<<(GN_B * GN_G + 255) / 256, 256, 0, stream>>>(ws, wgt, bias);

  const int rows_per_wave  = 16;
  const int rows_per_block = 8 * rows_per_wave;
  const int blocks3        = (N + rows_per_block - 1) / rows_per_block;
  gn_apply<<<blocks3, 256, 0, stream>>>(x, batch, ws, out, N, rows_per_wave);
}